// RecursiveSortNet_80479097192762
// MI455X (gfx1250) — compile-verified
//
#include <hip/hip_runtime.h>

typedef float v2f __attribute__((ext_vector_type(2)));
typedef float v4f __attribute__((ext_vector_type(4)));
typedef float v8f __attribute__((ext_vector_type(8)));

// ---------------------------------------------------------------------------
// Kernel 1: per-row 64x64 triple sort (axis1, axis2, axis1).
// Block = 64 threads (2 waves), one batch row per block.
// Each lane sorts one 64-element sequence ENTIRELY IN REGISTERS:
// compare-exchange = v_min_num + v_max_num (direction compile-time).
// LDS (64x65 padded tile) is used only to transpose between passes:
//   pass0: sequences = columns (loaded straight from global, coalesced)
//   pass1: sequences = rows
//   pass2: sequences = columns (stored straight to global, coalesced)
// Stride-65 padding -> both row reads (bank = (s+j)%64) and column reads
// (bank = (i+s)%64) are conflict-free. Only 2 barriers per matrix.
// ---------------------------------------------------------------------------
__device__ __forceinline__ void sort64_reg(float r[64]) {
#pragma unroll
  for (int k = 2; k <= 64; k <<= 1) {
#pragma unroll
    for (int j = k >> 1; j > 0; j >>= 1) {
#pragma unroll
      for (int q = 0; q < 32; ++q) {
        const int low = ((q & ~(j - 1)) << 1) | (q & (j - 1));
        const int hi  = low | j;
        const bool asc = (low & k) == 0;
        float a = r[low];
        float b = r[hi];
        float mn = fminf(a, b);
        float mx = fmaxf(a, b);
        r[low] = asc ? mn : mx;
        r[hi]  = asc ? mx : mn;
      }
    }
  }
}

__launch_bounds__(64)
__global__ void sort3_kernel(const float* __restrict__ x, float* __restrict__ xs) {
  __shared__ float tile[64 * 65];
  const int s = threadIdx.x;                    // sequence id 0..63
  const size_t base = (size_t)blockIdx.x * 4096;

  float r[64];
  for (int pass = 0; pass < 3; ++pass) {        // not unrolled: network emitted once
    // ---- load this pass's sequence ----
    if (pass == 0) {
#pragma unroll
      for (int e = 0; e < 64; ++e)              // column s of x: coalesced across lanes
        r[e] = x[base + (size_t)(e * 64 + s)];
    } else {
      __syncthreads();                          // wait for previous pass's LDS stores
      const int lb = (pass == 1) ? s * 65 : s;  // pass1: row s ; pass2: column s
      const int ls = (pass == 1) ? 1 : 65;
#pragma unroll
      for (int e = 0; e < 64; ++e)
        r[e] = tile[lb + e * ls];
    }

    sort64_reg(r);

    // ---- store this pass's sequence ----
    if (pass == 2) {
#pragma unroll
      for (int e = 0; e < 64; ++e)              // column s of xs: coalesced
        xs[base + (size_t)(e * 64 + s)] = r[e];
    } else {
      const int sb = (pass == 0) ? s : s * 65;  // pass0: column s ; pass1: row s
      const int ss = (pass == 0) ? 65 : 1;
#pragma unroll
      for (int e = 0; e < 64; ++e)
        tile[sb + e * ss] = r[e];
    }
  }
}

// ---------------------------------------------------------------------------
// GEMM: C[M,N] = act(A[M,K] @ W[N,K]^T + bias) via V_WMMA_F32_16X16X4_F32.
// Block: 256 threads (8 waves), tile 128(M) x 128(N), BK=16, double-buffered.
// Wave w: rows (w&3)*32, cols (w>>2)*64 -> 2x4 grid of 16x16 WMMA tiles
// (8 wmma per 6 ds_load_b64 per k-step).
// LDS layout (v2f = one k-pair, so every fragment is a single b64 op):
//   As2[row][k2], row stride 9 v2f (18 dwords, even -> 8B aligned)
//   Bs2[k2][n],   k2 stride 129 v2f
// Fragment mapping (ISA 7.12.2, f32):
//   A 16x4 : lane l holds A[l&15][2*(l>>4)+j]  (j=0,1) -> one v2f
//   B 4x16 : lane l holds B[2*(l>>4)+j][l&15]          -> one v2f
//   D 16x16: lane l holds D[(l>>4)*8+j][l&15]  (j=0..7)
// ---------------------------------------------------------------------------
template <bool RELU>
__launch_bounds__(256)
__global__ void gemm_wmma_kernel(const float* __restrict__ A,    // M x K
                                 const float* __restrict__ W,    // N x K
                                 const float* __restrict__ bias, // N
                                 float* __restrict__ C,          // M x N
                                 int M, int N, int K) {
  __shared__ v2f As2[2][128 * 9];
  __shared__ v2f Bs2[2][8 * 129];

  const int t    = threadIdx.x;
  const int lane = t & 31;
  const int w    = t >> 5;
  const int wm   = w & 3;              // 32-row group
  const int wn   = w >> 2;             // 64-col group
  const int lm   = lane & 15;
  const int lkh  = lane >> 4;          // k2 sub-offset {0,1}
  const int ld8  = (lane >> 4) << 3;   // D row sub-offset {0,8}

  const int n0 = blockIdx.x * 128;
  const int m0 = blockIdx.y * 128;

  const int srow = t >> 2;             // staging row 0..63 (x2 passes -> 128)
  const int skq  = (t & 3) << 2;       // staging k offset {0,4,8,12}
  const int sk2  = skq >> 1;

  v8f acc[2][4];
#pragma unroll
  for (int im = 0; im < 2; ++im)
#pragma unroll
    for (int in = 0; in < 4; ++in)
      acc[im][in] = (v8f)0.0f;

  v4f ra[2], rb[2];
  const int nch = K >> 4;

  // ---- prologue: stage chunk 0 ----
#pragma unroll
  for (int i = 0; i < 2; ++i) {
    ra[i] = *(const v4f*)(A + (size_t)(m0 + (i << 6) + srow) * K + skq);
    rb[i] = *(const v4f*)(W + (size_t)(n0 + (i << 6) + srow) * K + skq);
  }
#pragma unroll
  for (int i = 0; i < 2; ++i) {
    const int row = (i << 6) + srow;
    v2f alo; alo.x = ra[i].x; alo.y = ra[i].y;
    v2f ahi; ahi.x = ra[i].z; ahi.y = ra[i].w;
    As2[0][row * 9 + sk2]     = alo;
    As2[0][row * 9 + sk2 + 1] = ahi;
    v2f blo; blo.x = rb[i].x; blo.y = rb[i].y;
    v2f bhi; bhi.x = rb[i].z; bhi.y = rb[i].w;
    Bs2[0][sk2 * 129 + row]       = blo;
    Bs2[0][(sk2 + 1) * 129 + row] = bhi;
  }
  __syncthreads();

  for (int c = 0; c < nch; ++c) {
    const int cur = c & 1;
    // ---- prefetch next chunk from global (overlaps with wmma below) ----
    if (c + 1 < nch) {
      const int k0 = (c + 1) << 4;
#pragma unroll
      for (int i = 0; i < 2; ++i) {
        ra[i] = *(const v4f*)(A + (size_t)(m0 + (i << 6) + srow) * K + k0 + skq);
        rb[i] = *(const v4f*)(W + (size_t)(n0 + (i << 6) + srow) * K + k0 + skq);
      }
    }
    // ---- compute on current buffer: 4 k-steps x 8 wmma ----
#pragma unroll
    for (int ks = 0; ks < 4; ++ks) {
      const int k2 = (ks << 1) + lkh;
      v2f af[2], bf[4];
#pragma unroll
      for (int im = 0; im < 2; ++im)
        af[im] = As2[cur][((wm << 5) + (im << 4) + lm) * 9 + k2];
#pragma unroll
      for (int in = 0; in < 4; ++in)
        bf[in] = Bs2[cur][k2 * 129 + (wn << 6) + (in << 4) + lm];
#pragma unroll
      for (int im = 0; im < 2; ++im)
#pragma unroll
        for (int in = 0; in < 4; ++in)
          acc[im][in] = __builtin_amdgcn_wmma_f32_16x16x4_f32(
              /*neg_a=*/false, af[im], /*neg_b=*/false, bf[in],
              /*c_mod=*/(short)0, acc[im][in],
              /*reuse_a=*/false, /*reuse_b=*/false);
    }
    // ---- stage next chunk into the other buffer ----
    if (c + 1 < nch) {
      const int nxt = cur ^ 1;
#pragma unroll
      for (int i = 0; i < 2; ++i) {
        const int row = (i << 6) + srow;
        v2f alo; alo.x = ra[i].x; alo.y = ra[i].y;
        v2f ahi; ahi.x = ra[i].z; ahi.y = ra[i].w;
        As2[nxt][row * 9 + sk2]     = alo;
        As2[nxt][row * 9 + sk2 + 1] = ahi;
        v2f blo; blo.x = rb[i].x; blo.y = rb[i].y;
        v2f bhi; bhi.x = rb[i].z; bhi.y = rb[i].w;
        Bs2[nxt][sk2 * 129 + row]       = blo;
        Bs2[nxt][(sk2 + 1) * 129 + row] = bhi;
      }
    }
    __syncthreads();
  }

  // ---- epilogue: bias (+ReLU), 16 consecutive cols per half-wave ----
#pragma unroll
  for (int in = 0; in < 4; ++in) {
    const int col = n0 + (wn << 6) + (in << 4) + lm;
    const float bv = bias[col];
#pragma unroll
    for (int im = 0; im < 2; ++im) {
#pragma unroll
      for (int jj = 0; jj < 8; ++jj) {
        const int row = m0 + (wm << 5) + (im << 4) + ld8 + jj;
        float v = acc[im][in][jj] + bv;
        if (RELU) v = fmaxf(v, 0.0f);
        C[(size_t)row * N + col] = v;
      }
    }
  }
}

// ---------------------------------------------------------------------------
// Launch: sort -> (xs in d_out) -> GEMM1(relu) -> GEMM2(relu) -> GEMM3.
// xs reuses d_out (dead before GEMM3 overwrites it); h1/h2 live in d_ws.
// ---------------------------------------------------------------------------
extern "C" void kernel_launch(void* const* d_in, const int* in_sizes, int n_in,
                              void* d_out, int out_size, void* d_ws, size_t ws_size,
                              hipStream_t stream) {
  (void)in_sizes; (void)n_in; (void)out_size; (void)ws_size;
  const float* x  = (const float*)d_in[0];
  const float* W1 = (const float*)d_in[1];
  const float* b1 = (const float*)d_in[2];
  const float* W2 = (const float*)d_in[3];
  const float* b2 = (const float*)d_in[4];
  const float* W3 = (const float*)d_in[5];
  const float* b3 = (const float*)d_in[6];
  float* out = (float*)d_out;

  const int B = 8192, D = 4096, H1 = 512, H2 = 256;

  float* xs = out;                       // reuse output buffer (128 MB saved)
  float* h1 = (float*)d_ws;              // B x 512  (16 MB)
  float* h2 = h1 + (size_t)B * H1;       // B x 256  ( 8 MB)

  sort3_kernel<<<B, 64, 0, stream>>>(x, xs);

  gemm_wmma_kernel<true ><<<dim3(H1 / 128, B / 128), 256, 0, stream>>>(
      xs, W1, b1, h1, B, H1, D);
  gemm_wmma_kernel<true ><<<dim3(H2 / 128, B / 128), 256, 0, stream>>>(
      h1, W2, b2, h2, B, H2, H1);
  gemm_wmma_kernel<false><<<dim3(D  / 128, B / 128), 256, 0, stream>>>(
      h2, W3, b3, out, B, D, H2);
}